// SPDTangentSpace_79697413144763
// MI455X (gfx1250) — compile-verified
//
#include <hip/hip_runtime.h>
#include <math.h>

// CDNA5 / gfx1250: wave32, WMMA f32 16x16x4.
typedef float v2f __attribute__((ext_vector_type(2)));
typedef float v8f __attribute__((ext_vector_type(8)));

#define MAT_N   64
#define LDSZ    68          // padded LDS row stride (floats): (m*68+k)%64 = 4m+k -> conflict-free
#define TPB     256         // 8 waves of 32
#define NLEVEL  4           // square roots: log M = 2^NLEVEL * log(M^(1/2^NLEVEL))
#define MTERMS  7           // Mercator series terms for log(I+E)

// GEMM epilogue modes
#define EP_STORE  0   // C = acc  (acc pre-seeded with diag_init on the diagonal)
#define EP_AFFINE 1   // C = 1.5 I - 0.5 acc      (Newton-Schulz polynomial, fused)
#define EP_TRIU   2   // global out[triu(r,c)] = scale*acc + logc*I  (final, fused)

// C = A * B (+ fused epilogue) for 64x64 f32 matrices in LDS (stride LDSZ).
// 8 waves: wave wv owns rows [16*(wv>>1), +16) and column half (wv&1)*32 (2 tiles).
template <int EP>
__device__ __forceinline__ void gemm64(float* __restrict__ C,
                                       const float* __restrict__ A,
                                       const float* __restrict__ B,
                                       int lane, int wv,
                                       float diag_init, float scale, float logc) {
  const int m     = lane & 15;      // M index (A,C) / N index (B,C) per ISA layouts
  const int hi    = lane >> 4;      // half-wave select
  const int kk    = hi * 2;         // K split: lanes 0-15 -> K0/K1, lanes 16-31 -> K2/K3
  const int row0  = (wv >> 1) * 16;
  const int nbase = (wv & 1) * 32;
#pragma unroll
  for (int nt = 0; nt < 2; ++nt) {
    const int n0 = nbase + nt * 16;
    v8f acc;
#pragma unroll
    for (int v = 0; v < 8; ++v) {  // seed accumulator (diag_init * I)
      const int r = row0 + v + hi * 8;
      acc[v] = (r == (n0 + m)) ? diag_init : 0.0f;
    }
#pragma unroll
    for (int k0 = 0; k0 < MAT_N; k0 += 4) {
      v2f a, b;
      a.x = A[(row0 + m) * LDSZ + (k0 + kk)];
      a.y = A[(row0 + m) * LDSZ + (k0 + kk + 1)];
      b.x = B[(k0 + kk) * LDSZ + (n0 + m)];
      b.y = B[(k0 + kk + 1) * LDSZ + (n0 + m)];
      acc = __builtin_amdgcn_wmma_f32_16x16x4_f32(false, a, false, b,
                                                  (short)0, acc, false, false);
    }
#pragma unroll
    for (int v = 0; v < 8; ++v) {
      const int r = row0 + v + hi * 8;
      const int c = n0 + m;
      if (EP == EP_STORE) {
        C[r * LDSZ + c] = acc[v];
      } else if (EP == EP_AFFINE) {
        C[r * LDSZ + c] = ((r == c) ? 1.5f : 0.0f) - 0.5f * acc[v];
      } else { // EP_TRIU: C is the global output base for this matrix
        if (c >= r) {
          const int idx = r * MAT_N - (r * (r - 1)) / 2 + (c - r);
          C[idx] = scale * acc[v] + ((r == c) ? logc : 0.0f);
        }
      }
    }
  }
}

__global__ __launch_bounds__(TPB)
void spd_logm_kernel(const float* __restrict__ x, float* __restrict__ out) {
  __shared__ float buf0[MAT_N * LDSZ];
  __shared__ float buf1[MAT_N * LDSZ];
  __shared__ float buf2[MAT_N * LDSZ];
  __shared__ float buf3[MAT_N * LDSZ];
  __shared__ float red[MAT_N];
  __shared__ float sc[2];

  const int tid  = threadIdx.x;
  const int lane = tid & 31;
  const int wv   = tid >> 5;
  const int bid  = blockIdx.x;
  const float* A = x + (size_t)bid * MAT_N * MAT_N;

  float* pY = buf0;   // sqrt iterate
  float* pZ = buf1;   // inverse-sqrt partner / later E
  float* pT = buf2;   // NS polynomial / Horner accumulator
  float* pS = buf3;   // scratch

  // ---- load A into LDS ----
  for (int i = tid; i < MAT_N * MAT_N; i += TPB) {
    int r = i >> 6, c = i & 63;
    pY[r * LDSZ + c] = A[i];
  }
  __syncthreads();

  // ---- spectral bound c = ||A||_inf (>= lambda_max for SPD) ----
  if (tid < MAT_N) {
    float s = 0.f;
    for (int j = 0; j < MAT_N; ++j) s += fabsf(pY[tid * LDSZ + j]);
    red[tid] = s;
  }
  __syncthreads();
  if (tid == 0) {
    float c = red[0];
    for (int j = 1; j < MAT_N; ++j) c = fmaxf(c, red[j]);
    c = fmaxf(c, 1.0f);
    sc[0] = c;
    sc[1] = logf(c);
  }
  __syncthreads();
  const float cinv = 1.0f / sc[0];
  const float logc = sc[1];

  // ---- Y = A/c : spectrum in (0,1] ----
  for (int i = tid; i < MAT_N * MAT_N; i += TPB) {
    int r = i >> 6, c = i & 63;
    pY[r * LDSZ + c] *= cinv;
  }
  __syncthreads();

  // ---- NLEVEL repeated square roots, coupled Newton-Schulz (tapered) ----
  const int niter_lvl[NLEVEL] = {10, 8, 6, 5};
#pragma unroll 1
  for (int level = 0; level < NLEVEL; ++level) {
    // Iteration 0 has Z == I:  T = 1.5I - 0.5Y (elementwise), newY = Y*T, newZ = T.
    for (int i = tid; i < MAT_N * MAT_N; i += TPB) {
      int r = i >> 6, c = i & 63;
      pT[r * LDSZ + c] = ((r == c) ? 1.5f : 0.0f) - 0.5f * pY[r * LDSZ + c];
    }
    __syncthreads();
    gemm64<EP_STORE>(pS, pY, pT, lane, wv, 0.f, 0.f, 0.f);   // newY = Y*T
    __syncthreads();
    {
      float* oY = pY; float* oZ = pZ;
      pY = pS;   // newY
      pZ = pT;   // newZ = T
      pT = oY; pS = oZ;
    }
#pragma unroll 1
    for (int it = 1; it < niter_lvl[level]; ++it) {
      gemm64<EP_AFFINE>(pT, pZ, pY, lane, wv, 0.f, 0.f, 0.f); // T = 1.5I - 0.5*Z*Y
      __syncthreads();
      gemm64<EP_STORE>(pS, pY, pT, lane, wv, 0.f, 0.f, 0.f);  // newY = Y*T
      __syncthreads();
      gemm64<EP_STORE>(pY, pT, pZ, lane, wv, 0.f, 0.f, 0.f);  // newZ = T*Z
      __syncthreads();
      float* oldZ = pZ;
      pZ = pY;   // newZ
      pY = pS;   // newY
      pS = oldZ; // freed
    }
  }

  // ---- E = Y - I (into pZ); seed Horner P = c7*E + c6*I (into pT) ----
  const float c7 = 1.0f / 7.0f, c6 = -1.0f / 6.0f;
  for (int i = tid; i < MAT_N * MAT_N; i += TPB) {
    int r = i >> 6, c = i & 63;
    float e = pY[r * LDSZ + c] - ((r == c) ? 1.0f : 0.0f);
    pZ[r * LDSZ + c] = e;
    pT[r * LDSZ + c] = c7 * e + ((r == c) ? c6 : 0.0f);
  }
  __syncthreads();

  // ---- Horner: P <- E*P + c_j I, j = MTERMS-2 .. 1 (diag fused into WMMA C) ----
#pragma unroll 1
  for (int j = MTERMS - 2; j >= 1; --j) {
    const float cj = ((j & 1) ? 1.0f : -1.0f) / (float)j;
    gemm64<EP_STORE>(pS, pZ, pT, lane, wv, cj, 0.f, 0.f);     // S = E*P + cj*I
    __syncthreads();
    float* t = pT; pT = pS; pS = t;
  }

  // ---- final: log(I+E) = E*P; write 2^NLEVEL * (.) + logc*I, upper triangle ----
  const float scale = (float)(1 << NLEVEL);
  float* obase = out + (size_t)bid * (MAT_N * (MAT_N + 1) / 2);
  gemm64<EP_TRIU>(obase, pZ, pT, lane, wv, 0.f, scale, logc);
}

extern "C" void kernel_launch(void* const* d_in, const int* in_sizes, int n_in,
                              void* d_out, int out_size, void* d_ws, size_t ws_size,
                              hipStream_t stream) {
  const float* x = (const float*)d_in[0];
  float* out = (float*)d_out;
  const int nb = in_sizes[0] / (MAT_N * MAT_N);   // 8192 matrices
  (void)n_in; (void)out_size; (void)d_ws; (void)ws_size;
  spd_logm_kernel<<<nb, TPB, 0, stream>>>(x, out);
}